// GPUPacketLoss_8323646620198
// MI455X (gfx1250) — compile-verified
//
#include <hip/hip_runtime.h>
#include <stdint.h>

// Problem constants (match reference)
#define B_DIM 128
#define T_DIM 160000
#define NSEG  200
#define BLK   256
#define CHUNKS 8
#define TILE (CHUNKS * BLK * 4)   // 8192 samples per block tile

typedef float v4f __attribute__((ext_vector_type(4)));

// ---------------------------------------------------------------------------
// Kernel 1: per-row segment construction (exact fp32 semantics of reference)
// ---------------------------------------------------------------------------
__global__ __launch_bounds__(BLK) void seg_build_kernel(
    const float* __restrict__ ratio,
    const int*   __restrict__ seg_len_raw,
    const float* __restrict__ start_u,
    int2*        __restrict__ segs)
{
  const int b   = blockIdx.x;
  const int tid = threadIdx.x;

  __shared__ int s_scan[BLK];
  const int raw = (tid < NSEG) ? seg_len_raw[b * NSEG + tid] : 0;
  s_scan[tid] = raw;
  __syncthreads();

  // Hillis-Steele inclusive scan over 256 entries (200 valid, rest 0)
  #pragma unroll
  for (int off = 1; off < BLK; off <<= 1) {
    int v = (tid >= off) ? s_scan[tid - off] : 0;
    __syncthreads();
    s_scan[tid] += v;
    __syncthreads();
  }

  if (tid < NSEG) {
    const int prev  = s_scan[tid] - raw;                       // exclusive prefix
    const int total = (int)floorf((float)T_DIM * ratio[b]);    // fp32 RNE mul, floor
    int eff = total - prev;
    eff = eff < 0 ? 0 : eff;
    eff = eff > raw ? raw : eff;                               // clip(., 0, raw)
    const float u = start_u[b * NSEG + tid];
    const int start = (int)floorf(u * (float)(T_DIM - eff + 1)); // exact: 160001 < 2^24
    segs[b * NSEG + tid] = make_int2(start, start + eff);      // start==end when eff==0
  }
}

// ---------------------------------------------------------------------------
// Kernel 2: streaming mask-apply. One block = one 8192-sample tile of one row.
// Segments staged to LDS via CDNA5 async global->LDS DMA; audio streamed with
// prefetch + non-temporal hints (single-pass data, keep L2 for segment tables).
// ---------------------------------------------------------------------------
__global__ __launch_bounds__(BLK) void apply_kernel(
    const float* __restrict__ audio,
    const int2*  __restrict__ segs,
    float*       __restrict__ out)
{
  const int b        = blockIdx.y;
  const int tileBase = blockIdx.x * TILE;
  const int tid      = threadIdx.x;

  const float* arow = audio + (size_t)b * T_DIM;
  float*       orow = out   + (size_t)b * T_DIM;

  // Kick speculative prefetch of our streaming tile (global_prefetch_b8)
  // while the async segment DMA is in flight.
  #pragma unroll
  for (int c = 0; c < CHUNKS; ++c) {
    int p = tileBase + c * (BLK * 4) + tid * 4;
    if (p < T_DIM) __builtin_prefetch(arow + p, 0, 0);
  }

  __shared__ int2 s_all[BLK];   // raw staged pairs (lanes >=200 hold a dup)
  __shared__ int2 s_flt[NSEG];  // segments overlapping this tile
  __shared__ int  s_cnt;
  if (tid == 0) s_cnt = 0;

  // Async global -> LDS copy of this row's 200 (start,end) pairs, 8B per lane.
  {
    const int sidx = (tid < NSEG) ? tid : (NSEG - 1);
    const int2* gp = segs + (size_t)b * NSEG + sidx;
    unsigned long long ga = (unsigned long long)(uintptr_t)gp;
    unsigned loff = (unsigned)(uintptr_t)(&s_all[tid]); // low 32 bits = LDS offset
    asm volatile("global_load_async_to_lds_b64 %0, %1, off"
                 :: "v"(loff), "v"(ga) : "memory");
    asm volatile("s_wait_asynccnt 0" ::: "memory");
  }
  __syncthreads();

  // Filter segments overlapping [tileBase, tileEnd); empty segments excluded.
  const int tileEnd = (tileBase + TILE < T_DIM) ? (tileBase + TILE) : T_DIM;
  if (tid < NSEG) {
    int2 se = s_all[tid];
    if (se.y > se.x && se.x < tileEnd && se.y > tileBase) {
      int k = atomicAdd(&s_cnt, 1);
      s_flt[k] = se;
    }
  }
  __syncthreads();
  const int n = s_cnt;   // typically ~4 segments per 8192-sample tile

  // Stream 8 float4s per thread: load NT, mask via interval clamp, store NT.
  #pragma unroll
  for (int c = 0; c < CHUNKS; ++c) {
    int p = tileBase + c * (BLK * 4) + tid * 4;
    if (p < T_DIM) {                       // T%4==0 -> whole float4 in range
      v4f v = __builtin_nontemporal_load((const v4f*)(arow + p));
      unsigned m = 0;
      for (int k = 0; k < n; ++k) {
        int lo = s_flt[k].x - p;
        int hi = s_flt[k].y - p;
        lo = lo < 0 ? 0 : lo;
        hi = hi > 4 ? 4 : hi;
        if (hi > lo) m |= ((1u << hi) - (1u << lo));
      }
      if (m & 1u) v.x = 0.0f;
      if (m & 2u) v.y = 0.0f;
      if (m & 4u) v.z = 0.0f;
      if (m & 8u) v.w = 0.0f;
      __builtin_nontemporal_store(v, (v4f*)(orow + p));
    }
  }
}

// ---------------------------------------------------------------------------
extern "C" void kernel_launch(void* const* d_in, const int* in_sizes, int n_in,
                              void* d_out, int out_size, void* d_ws, size_t ws_size,
                              hipStream_t stream) {
  (void)in_sizes; (void)n_in; (void)out_size; (void)ws_size;

  const float* audio       = (const float*)d_in[0];   // (B, T) f32
  const float* ratio       = (const float*)d_in[1];   // (B,)   f32
  const int*   seg_len_raw = (const int*)  d_in[2];   // (B, NSEG) i32
  const float* start_u     = (const float*)d_in[3];   // (B, NSEG) f32
  float*       out         = (float*)d_out;           // (B, T) f32
  int2*        segs        = (int2*)d_ws;             // B*NSEG*8 = 200 KiB scratch

  seg_build_kernel<<<B_DIM, BLK, 0, stream>>>(ratio, seg_len_raw, start_u, segs);

  dim3 grid((T_DIM + TILE - 1) / TILE, B_DIM);        // 20 x 128 blocks
  apply_kernel<<<grid, BLK, 0, stream>>>(audio, segs, out);
}